// SpectralPrototypeConsistencyLoss_65738769433118
// MI455X (gfx1250) — compile-verified
//
#include <hip/hip_runtime.h>
#include <cstdint>
#include <cmath>

// Problem constants (from the reference): B=2, C=64, H=W=D=64, K=2
#define NVOX        (64 * 64 * 64)   // voxels per batch = 262144
#define CCH         64               // channels
#define NB          2                // batches
#define NK          2                // classes
#define TPB         256              // threads per block (8 wave32s)
#define VPT         4                // voxels per thread (float4 / B128)
#define VOX_PER_BLK (TPB * VPT)      // 1024 voxels per block
#define BLKS_PER_B  (NVOX / VOX_PER_BLK)  // 256
#define CH_STRIDE_B ((size_t)NVOX * 4)    // byte stride between channel planes (1 MB)
#define PIPE        4                // LDS ring buffers
#define DEPTH       3                // async loads in flight per wave

// ---- CDNA5 async global->LDS copy (per-lane B128), tracked by ASYNCcnt ----
// Non-temporal: the 134 MB feature stream is touched exactly once.
__device__ __forceinline__ void async_load_b128(uintptr_t gaddr, unsigned lds_byte_off) {
    asm volatile("global_load_async_to_lds_b128 %0, %1, off th:TH_LOAD_NT"
                 :
                 : "v"(lds_byte_off), "v"((unsigned long long)gaddr)
                 : "memory");
}
#define WAIT_ASYNCCNT(N) asm volatile("s_wait_asynccnt %0" :: "n"(N) : "memory")

__global__ void __launch_bounds__(TPB)
spcl_main(const float* __restrict__ features,   // (B, C, N) fp32
          const int*   __restrict__ targets,    // (B, 1, N) int32 (values 0..K-1)
          const float* __restrict__ prototypes, // (K, C) fp32
          float*       __restrict__ ws)         // [B*K sums | B*K counts]
{
    __shared__ __align__(16) float featBuf[PIPE][VOX_PER_BLK]; // 16 KB ring
    __shared__ __align__(16) float protoI[NK * CCH];           // interleaved {p0[c],p1[c]}
    __shared__ float red[4];                                   // s0, s1, c0, c1

    const int tid = threadIdx.x;
    const int blk = blockIdx.x;
    const int b   = blk / BLKS_PER_B;
    const int n0  = (blk % BLKS_PER_B) * VOX_PER_BLK;

    // Interleave prototype rows: protoI[2c] = p0[c], protoI[2c+1] = p1[c]
    if (tid < CCH) {
        protoI[2 * tid]     = prototypes[tid];
        protoI[2 * tid + 1] = prototypes[CCH + tid];
    }
    if (tid < 4) red[tid] = 0.0f;

    // Per-thread class labels for its 4 voxels.
    const int4 t = *reinterpret_cast<const int4*>(
        targets + (size_t)b * NVOX + n0 + (size_t)tid * VPT);

    // Per-thread global base for channel 0, and LDS destination slot.
    const uintptr_t gbase =
        (uintptr_t)(features + ((size_t)b * CCH) * NVOX + n0 + (size_t)tid * VPT);
    const unsigned ldsThread =
        (unsigned)(uintptr_t)(&featBuf[0][0]) + (unsigned)tid * (VPT * 4u);

    // Prime the async pipeline: channels 0..DEPTH-1 into ring slots 0..DEPTH-1.
#pragma unroll
    for (int p = 0; p < DEPTH; ++p)
        async_load_b128(gbase + (size_t)p * CH_STRIDE_B,
                        ldsThread + (unsigned)(p & (PIPE - 1)) * (VOX_PER_BLK * 4u));

    __syncthreads();  // protoI / red visible block-wide

    float a0 = 0.0f, a1 = 0.0f, a2 = 0.0f, a3 = 0.0f;

    auto body = [&](int c) {
        const float4 f = *reinterpret_cast<const float4*>(&featBuf[c & (PIPE - 1)][tid * VPT]);
        const float2 p = *reinterpret_cast<const float2*>(&protoI[2 * c]); // 1x ds_load_b64, broadcast
        float pc, d;
        pc = (t.x != 0) ? p.y : p.x; d = f.x - pc; a0 = fmaf(d, d, a0);
        pc = (t.y != 0) ? p.y : p.x; d = f.y - pc; a1 = fmaf(d, d, a1);
        pc = (t.z != 0) ? p.y : p.x; d = f.z - pc; a2 = fmaf(d, d, a2);
        pc = (t.w != 0) ? p.y : p.x; d = f.w - pc; a3 = fmaf(d, d, a3);
    };

    // Steady state: keep DEPTH loads in flight; in-order completion means
    // ASYNCcnt <= DEPTH-1 guarantees the oldest tile has landed in LDS.
    for (int c = 0; c < CCH - DEPTH; ++c) {
        WAIT_ASYNCCNT(DEPTH - 1);
        body(c);
        async_load_b128(gbase + (size_t)(c + DEPTH) * CH_STRIDE_B,
                        ldsThread + (unsigned)((c + DEPTH) & (PIPE - 1)) * (VOX_PER_BLK * 4u));
    }
    // Drain: everything outstanding must be in LDS.
    WAIT_ASYNCCNT(0);
#pragma unroll
    for (int c = CCH - DEPTH; c < CCH; ++c) body(c);

    // Per-voxel distance, per-thread per-class partials.
    const float d0 = sqrtf(a0), d1 = sqrtf(a1), d2 = sqrtf(a2), d3 = sqrtf(a3);
    float s0 = 0.0f, s1 = 0.0f, c0 = 0.0f, c1 = 0.0f;
    if (t.x != 0) { s1 += d0; c1 += 1.0f; } else { s0 += d0; c0 += 1.0f; }
    if (t.y != 0) { s1 += d1; c1 += 1.0f; } else { s0 += d1; c0 += 1.0f; }
    if (t.z != 0) { s1 += d2; c1 += 1.0f; } else { s0 += d2; c0 += 1.0f; }
    if (t.w != 0) { s1 += d3; c1 += 1.0f; } else { s0 += d3; c0 += 1.0f; }

    // Block reduction in LDS (ds_add_f32), then 4 global fp32 atomics per block.
    atomicAdd(&red[0], s0);
    atomicAdd(&red[1], s1);
    atomicAdd(&red[2], c0);
    atomicAdd(&red[3], c1);
    __syncthreads();
    if (tid < 4) {
        const int gi = (tid < 2) ? (b * NK + tid)                  // sums
                                 : (NB * NK + b * NK + (tid - 2)); // counts
        unsafeAtomicAdd(&ws[gi], red[tid]);
    }
}

__global__ void __launch_bounds__(TPB)
spcl_zero_ws(float* __restrict__ ws) {
    if (blockIdx.x == 0 && threadIdx.x < 2 * NB * NK)  // 4 sums + 4 counts
        ws[threadIdx.x] = 0.0f;
}

__global__ void __launch_bounds__(32)
spcl_finalize(const float* __restrict__ ws, float* __restrict__ out) {
    if (blockIdx.x == 0 && threadIdx.x == 0) {
        float tot = 0.0f, nvalid = 0.0f;
        for (int i = 0; i < NB * NK; ++i) {
            const float cnt = ws[NB * NK + i];
            if (cnt > 0.0f) { tot += ws[i] / cnt; nvalid += 1.0f; }
        }
        out[0] = (nvalid > 0.0f) ? (tot / nvalid) : 0.0f;
    }
}

extern "C" void kernel_launch(void* const* d_in, const int* in_sizes, int n_in,
                              void* d_out, int out_size, void* d_ws, size_t ws_size,
                              hipStream_t stream) {
    // setup_inputs order: features, predictions (unused), targets, prototypes
    const float* features   = (const float*)d_in[0];
    const int*   targets    = (const int*)d_in[2];
    const float* prototypes = (const float*)d_in[3];
    float* ws  = (float*)d_ws;
    float* out = (float*)d_out;

    spcl_zero_ws<<<1, TPB, 0, stream>>>(ws);
    spcl_main<<<NB * BLKS_PER_B, TPB, 0, stream>>>(features, targets, prototypes, ws);
    spcl_finalize<<<1, 32, 0, stream>>>(ws, out);
}